// Set2Set_59760174957060
// MI455X (gfx1250) — compile-verified
//
#include <hip/hip_runtime.h>
#include <math.h>

#define IN    128
#define B2    256          // 2*IN
#define KCAT  384          // 2*IN + IN  (concat [q | r | h])
#define GATES 512          // 4*IN
#define BATCH 1024
#define STEPS 3
#define EMAX  4096         // LDS-cached scores per segment (overflow -> global)

typedef float v2f __attribute__((ext_vector_type(2)));
typedef float v8f __attribute__((ext_vector_type(8)));

__device__ __forceinline__ float sigmoidf_(float x) { return 1.0f / (1.0f + expf(-x)); }

__device__ __forceinline__ float wave_sum32(float p) {
    p += __shfl_xor(p, 16);
    p += __shfl_xor(p, 8);
    p += __shfl_xor(p, 4);
    p += __shfl_xor(p, 2);
    p += __shfl_xor(p, 1);
    return p;
}

// ---------------------------------------------------------------------------
// Segment offsets from sorted batch vector: seg[b] = first index of segment b,
// seg[BATCH] = N.  Handles empty segments.
// ---------------------------------------------------------------------------
__global__ void k_segstart(const int* __restrict__ batch, int* __restrict__ seg, int N) {
    int n = blockIdx.x * blockDim.x + threadIdx.x;
    if (n >= N) return;
    int b    = batch[n];
    int prev = (n == 0) ? -1 : batch[n - 1];
    for (int bb = prev + 1; bb <= b; ++bb) seg[bb] = n;
    if (n == N - 1)
        for (int bb = b + 1; bb <= BATCH; ++bb) seg[bb] = N;
}

// ---------------------------------------------------------------------------
// WcatT[k][n] = k<256 ? w_ih[n][k] : w_hh[n][k-256];  bsum[n] = b_ih[n]+b_hh[n]
// ---------------------------------------------------------------------------
__global__ void k_prepw(const float* __restrict__ w_ih, const float* __restrict__ w_hh,
                        const float* __restrict__ b_ih, const float* __restrict__ b_hh,
                        float* __restrict__ WT, float* __restrict__ bsum) {
    int t = blockIdx.x * blockDim.x + threadIdx.x;
    if (t < GATES) bsum[t] = b_ih[t] + b_hh[t];
    if (t >= GATES * KCAT) return;
    int n = t % GATES;
    int k = t / GATES;
    WT[t] = (k < B2) ? w_ih[(size_t)n * B2 + k] : w_hh[(size_t)n * IN + (k - B2)];
}

// ---------------------------------------------------------------------------
// gates[1024x512] = Acat[1024x384] @ WcatT[384x512] via V_WMMA_F32_16X16X4_F32.
// One wave per 16x16 tile; 64x32 = 2048 tiles.
// ---------------------------------------------------------------------------
__global__ void k_gates_wmma(const float* __restrict__ Acat, const float* __restrict__ WT,
                             float* __restrict__ gates) {
    int wave = (blockIdx.x * blockDim.x + threadIdx.x) >> 5;   // 0..2047
    int lane = threadIdx.x & 31;
    int tm   = wave >> 5;                                      // 0..63
    int tn   = wave & 31;                                      // 0..31
    int m0 = tm * 16, n0 = tn * 16;
    bool hi = lane >= 16;
    int  l16 = lane & 15;
    int  ka  = hi ? 2 : 0;

    const float* arow = Acat + (size_t)(m0 + l16) * KCAT + ka;     // even idx: 8B aligned
    const float* bcol = WT + (size_t)ka * GATES + (n0 + l16);

    v8f acc = {0.f, 0.f, 0.f, 0.f, 0.f, 0.f, 0.f, 0.f};
    #pragma unroll 8
    for (int k0 = 0; k0 < KCAT; k0 += 4) {
        float2 av = *(const float2*)(arow + k0);
        v2f a; a.x = av.x; a.y = av.y;
        v2f b;
        b.x = bcol[(size_t)k0 * GATES];
        b.y = bcol[(size_t)(k0 + 1) * GATES];
        acc = __builtin_amdgcn_wmma_f32_16x16x4_f32(
                  false, a, false, b, (short)0, acc, false, false);
    }

    float* gout = gates + (size_t)(m0 + (hi ? 8 : 0)) * GATES + (n0 + l16);
    #pragma unroll
    for (int v = 0; v < 8; ++v) gout[(size_t)v * GATES] = acc[v];
}

// ---------------------------------------------------------------------------
// LSTM pointwise (gate order i,f,g,o); writes h into q-slot and h-slot of Acat.
// ---------------------------------------------------------------------------
__global__ void k_lstm_pw(const float* __restrict__ gates, const float* __restrict__ bsum,
                          float* __restrict__ c, float* __restrict__ Acat) {
    int t = blockIdx.x * blockDim.x + threadIdx.x;
    if (t >= BATCH * IN) return;
    int b = t >> 7, j = t & (IN - 1);
    const float* g = gates + (size_t)b * GATES;
    float gi = g[j]          + bsum[j];
    float gf = g[IN + j]     + bsum[IN + j];
    float gg = g[2 * IN + j] + bsum[2 * IN + j];
    float go = g[3 * IN + j] + bsum[3 * IN + j];
    float cc = c[t];
    cc = sigmoidf_(gf) * cc + sigmoidf_(gi) * tanhf(gg);
    float hh = sigmoidf_(go) * tanhf(cc);
    c[t] = cc;
    float* ar = Acat + (size_t)b * KCAT;
    ar[j]          = hh;   // q part of q_star
    ar[2 * IN + j] = hh;   // h input for next LSTM step
}

// ---------------------------------------------------------------------------
// Fused attention: pass A (scores + segment max) and pass B (softmax-weighted
// readout) in ONE block per segment, so the second x read hits L2 (192MB)
// while the segment (~250KB) is still resident.  Scores cached in LDS.
// 8 waves/block; wave w owns nodes s+w, s+w+8, ...; two nodes per iteration.
// ---------------------------------------------------------------------------
__global__ void k_attn(const float* __restrict__ x, float* __restrict__ Acat,
                       const int* __restrict__ seg, float* __restrict__ eglob) {
    __shared__ float se[EMAX];
    __shared__ float smax[8];
    __shared__ float sred[8 * IN];
    __shared__ float sden[8];
    __shared__ float sbc[2];           // [0]=segment max, [1]=denominator

    int b    = blockIdx.x;
    int lane = threadIdx.x & 31;
    int wave = threadIdx.x >> 5;
    int s = seg[b], en = seg[b + 1];

    float4 q4 = *(const float4*)(Acat + (size_t)b * KCAT + 4 * lane);

    // ---- pass A: e[n] = x[n].q ; segment max --------------------------------
    float wmax = -INFINITY;
    for (int n = s + wave; n < en; n += 16) {
        int n2 = n + 8;
        bool has2 = n2 < en;
        const float* xr1 = x + (size_t)n * IN + 4 * lane;
        float4 xa = *(const float4*)xr1;
        float4 xb = {0.f, 0.f, 0.f, 0.f};
        if (has2) xb = *(const float4*)(x + (size_t)n2 * IN + 4 * lane);
        if (n + 16 < en) __builtin_prefetch(xr1 + 16 * IN, 0, 0);

        float p = wave_sum32(xa.x * q4.x + xa.y * q4.y + xa.z * q4.z + xa.w * q4.w);
        if (lane == 0) { int i = n - s; if (i < EMAX) se[i] = p; else eglob[n] = p; }
        wmax = fmaxf(wmax, p);
        if (has2) {
            float p2 = wave_sum32(xb.x * q4.x + xb.y * q4.y + xb.z * q4.z + xb.w * q4.w);
            if (lane == 0) { int i = n2 - s; if (i < EMAX) se[i] = p2; else eglob[n2] = p2; }
            wmax = fmaxf(wmax, p2);
        }
    }
    if (lane == 0) smax[wave] = wmax;
    __syncthreads();
    if (threadIdx.x == 0) {
        float mm = smax[0];
        #pragma unroll
        for (int i = 1; i < 8; ++i) mm = fmaxf(mm, smax[i]);
        sbc[0] = mm;
    }
    __syncthreads();
    float mb = sbc[0];

    // ---- pass B: denom + weighted sum (x re-read: L2-hot) -------------------
    float4 r4 = {0.f, 0.f, 0.f, 0.f};
    float  dacc = 0.f;
    for (int n = s + wave; n < en; n += 16) {
        int n2 = n + 8;
        bool has2 = n2 < en;
        const float* xr1 = x + (size_t)n * IN + 4 * lane;
        float4 xa = *(const float4*)xr1;
        float4 xb = {0.f, 0.f, 0.f, 0.f};
        if (has2) xb = *(const float4*)(x + (size_t)n2 * IN + 4 * lane);
        if (n + 16 < en) __builtin_prefetch(xr1 + 16 * IN, 0, 0);

        int i1 = n - s;
        float e1 = (i1 < EMAX) ? se[i1] : eglob[n];
        float ex = expf(e1 - mb);
        r4.x += ex * xa.x; r4.y += ex * xa.y; r4.z += ex * xa.z; r4.w += ex * xa.w;
        dacc += ex;                      // identical across lanes of this wave
        if (has2) {
            int i2 = n2 - s;
            float e2 = (i2 < EMAX) ? se[i2] : eglob[n2];
            float ex2 = expf(e2 - mb);
            r4.x += ex2 * xb.x; r4.y += ex2 * xb.y; r4.z += ex2 * xb.z; r4.w += ex2 * xb.w;
            dacc += ex2;
        }
    }
    float* wr = sred + wave * IN + 4 * lane;
    wr[0] = r4.x; wr[1] = r4.y; wr[2] = r4.z; wr[3] = r4.w;
    if (lane == 0) sden[wave] = dacc;
    __syncthreads();
    if (threadIdx.x == 0) {
        float dd = 0.f;
        #pragma unroll
        for (int i = 0; i < 8; ++i) dd += sden[i];
        sbc[1] = dd;
    }
    __syncthreads();
    int tid = threadIdx.x;
    if (tid < IN) {
        float sum = 0.f;
        #pragma unroll
        for (int w = 0; w < 8; ++w) sum += sred[w * IN + tid];
        float dd = sbc[1];
        Acat[(size_t)b * KCAT + IN + tid] = (dd > 0.f) ? sum / dd : 0.f;
    }
}

// ---------------------------------------------------------------------------
// q_star = Acat[:, 0:256]
// ---------------------------------------------------------------------------
__global__ void k_out(const float* __restrict__ Acat, float* __restrict__ out) {
    int t = blockIdx.x * blockDim.x + threadIdx.x;
    if (t >= BATCH * B2) return;
    int b = t >> 8, j = t & (B2 - 1);
    out[t] = Acat[(size_t)b * KCAT + j];
}

// ---------------------------------------------------------------------------
extern "C" void kernel_launch(void* const* d_in, const int* in_sizes, int n_in,
                              void* d_out, int out_size, void* d_ws, size_t ws_size,
                              hipStream_t stream) {
    const float* x     = (const float*)d_in[0];
    const float* w_ih  = (const float*)d_in[1];
    const float* w_hh  = (const float*)d_in[2];
    const float* b_ih  = (const float*)d_in[3];
    const float* b_hh  = (const float*)d_in[4];
    const int*   batch = (const int*)d_in[5];
    const int N = in_sizes[0] / IN;

    // workspace layout (floats)
    float* ws    = (float*)d_ws;
    float* Acat  = ws;                               // 1024*384
    float* c     = Acat  + (size_t)BATCH * KCAT;     // 1024*128
    float* WT    = c     + (size_t)BATCH * IN;       // 384*512
    float* bsum  = WT    + (size_t)KCAT * GATES;     // 512
    float* gates = bsum  + GATES;                    // 1024*512
    float* eglob = gates + (size_t)BATCH * GATES;    // N (overflow scores only)
    int*   seg   = (int*)(eglob + N);                // 1025

    // zero state (Acat and c are contiguous) -- every call, graph-safe
    hipMemsetAsync(Acat, 0, (size_t)(BATCH * KCAT + BATCH * IN) * sizeof(float), stream);

    k_segstart<<<(N + 255) / 256, 256, 0, stream>>>(batch, seg, N);
    k_prepw<<<(GATES * KCAT + 255) / 256, 256, 0, stream>>>(w_ih, w_hh, b_ih, b_hh, WT, bsum);

    for (int step = 0; step < STEPS; ++step) {
        k_gates_wmma<<<256, 256, 0, stream>>>(Acat, WT, gates);
        k_lstm_pw<<<(BATCH * IN) / 256, 256, 0, stream>>>(gates, bsum, c, Acat);
        k_attn<<<BATCH, 256, 0, stream>>>(x, Acat, seg, eglob);
    }

    k_out<<<(BATCH * B2) / 256, 256, 0, stream>>>(Acat, (float*)d_out);
}